// PtModule_73667279061069
// MI455X (gfx1250) — compile-verified
//
#include <hip/hip_runtime.h>

// y = (x + 1) * 2 / 3 ; where y > 0 -> y - 5
// Pure streaming elementwise op over 8192x8192 fp32: HBM-bound
// (512 MB traffic => ~22 us floor at 23.3 TB/s).
// Strategy: B128 non-temporal loads/stores, 4 independent loads in flight
// per lane on the fast path, 32-bit indexing so addressing uses the
// SGPR-base + 32-bit-VGPR-offset (GVS) mode instead of 64-bit VALU math.

typedef float v4f __attribute__((ext_vector_type(4)));

__device__ __forceinline__ v4f apply_op(v4f x) {
    const float C = 2.0f / 3.0f;            // compile-time constant
    v4f y = (x + 1.0f) * C;                 // v_add_f32 + v_mul_f32 per lane
    v4f r;
#pragma unroll
    for (int k = 0; k < 4; ++k) {
        float yk = y[k];
        r[k] = (yk > 0.0f) ? (yk - 5.0f) : yk;  // v_cmp + v_cndmask path
    }
    return r;
}

__global__ __launch_bounds__(256) void ewise_stream_kernel(
    const float* __restrict__ x, float* __restrict__ out, int n) {
    const int nvec   = n >> 2;                                  // float4 count
    const int stride = (int)(gridDim.x * blockDim.x);           // in float4s
    const int tid    = (int)(blockIdx.x * blockDim.x + threadIdx.x);

    const v4f* __restrict__ xv = (const v4f*)x;
    v4f* __restrict__ ov       = (v4f*)out;

    const int i0 = tid;
    const int i1 = tid + stride;
    const int i2 = tid + 2 * stride;
    const int i3 = tid + 3 * stride;

    if (i3 < nvec) {
        // Fast path: 4 NT B128 loads issued back-to-back (4 on LOADcnt),
        // then compute, then 4 NT B128 stores.
        v4f a0 = __builtin_nontemporal_load(&xv[i0]);
        v4f a1 = __builtin_nontemporal_load(&xv[i1]);
        v4f a2 = __builtin_nontemporal_load(&xv[i2]);
        v4f a3 = __builtin_nontemporal_load(&xv[i3]);

        v4f r0 = apply_op(a0);
        v4f r1 = apply_op(a1);
        v4f r2 = apply_op(a2);
        v4f r3 = apply_op(a3);

        __builtin_nontemporal_store(r0, &ov[i0]);
        __builtin_nontemporal_store(r1, &ov[i1]);
        __builtin_nontemporal_store(r2, &ov[i2]);
        __builtin_nontemporal_store(r3, &ov[i3]);
    } else {
        // Straggler path: guarded grid-stride over remaining float4s.
        for (int i = i0; i < nvec; i += stride) {
            v4f a = __builtin_nontemporal_load(&xv[i]);
            __builtin_nontemporal_store(apply_op(a), &ov[i]);
        }
    }

    // Scalar tail for n % 4 != 0 (not hit for 8192^2; kept for generality).
    const int tail_base = nvec << 2;
    const int t = tail_base + tid;
    if (t < n) {
        float v = x[t];
        float y = (v + 1.0f) * (2.0f / 3.0f);
        out[t] = (y > 0.0f) ? (y - 5.0f) : y;
    }
}

extern "C" void kernel_launch(void* const* d_in, const int* in_sizes, int n_in,
                              void* d_out, int out_size, void* d_ws, size_t ws_size,
                              hipStream_t stream) {
    (void)n_in; (void)d_ws; (void)ws_size; (void)out_size;
    const float* x = (const float*)d_in[0];
    float* out     = (float*)d_out;

    const int n    = in_sizes[0];        // 67,108,864 for 8192^2
    const int nvec = n >> 2;             // float4 units

    const int block = 256;               // 8 wave32 per block
    const int perB  = block * 4;         // 4 float4 per thread
    int grid        = (nvec + perB - 1) / perB;  // 16384 for this size
    if (grid < 1) grid = 1;

    ewise_stream_kernel<<<grid, block, 0, stream>>>(x, out, n);
}